// MCHyperConnectionRes_75685913690742
// MI455X (gfx1250) — compile-verified
//
#include <hip/hip_runtime.h>
#include <hip/hip_bf16.h>

typedef __attribute__((ext_vector_type(2))) float v2f;
typedef __attribute__((ext_vector_type(8))) float v8f;

// Problem constants (from reference setup_inputs): b=2, s=4096, n=4, d=2048
#define K_DIM   8192          // n*d
#define N_TOK   8192          // b*s
#define BK      64            // K-tile staged in LDS per step
#define LDS_STR 68            // padded row stride (floats): 272B rows, 16B aligned, conflict-free
#define TOK_PER_WAVE 16       // WMMA M
#define WAVES_PER_BLK 4
#define TOK_PER_BLK (TOK_PER_WAVE * WAVES_PER_BLK)   // 64
#define N_KTILES (K_DIM / BK)                        // 128

// global -> LDS async copy of 16 bytes (per lane), tracked by ASYNCcnt.
__device__ __forceinline__ void async_cp16(const float* g, const float* lds) {
    const unsigned long long ga = (unsigned long long)(size_t)g;
    const unsigned la = (unsigned)(size_t)lds;   // low 32 bits of generic shared ptr = LDS offset
    asm volatile("global_load_async_to_lds_b128 %0, %1, off"
                 :: "v"(la), "v"(ga) : "memory");
}

// ---------------------------------------------------------------------------
// Kernel 1: logits = x_norm @ W^T via V_WMMA_F32_16X16X4_F32 with
// double-buffered async global->LDS staging, then h = exp(logits*factor+bias)
// and 20 Sinkhorn iterations via shfl_xor in the WMMA C-layout.
// ---------------------------------------------------------------------------
__global__ __launch_bounds__(128) void MCHyper_logits_sinkhorn_kernel(
    const float* __restrict__ xnorm,   // (N_TOK, K_DIM)
    const float* __restrict__ W,       // (16, K_DIM)
    const float* __restrict__ factor,  // (1,)
    const float* __restrict__ bias,    // (16,) row-major 4x4
    float* __restrict__ h_out)         // (N_TOK, 16)
{
    __shared__ float As[2][TOK_PER_BLK * LDS_STR];  // 2 x 64 x 68 floats
    __shared__ float Ws[2][16 * LDS_STR];           // 2 x 16 x 68 floats

    const int tid    = threadIdx.x;
    const int wave   = tid >> 5;
    const int lane   = tid & 31;
    const int row16  = lane & 15;          // M (A) / N (B) index
    const int khalf  = (lane >> 4) << 1;   // 0 or 2: K-pair selected by lane half
    const int tokBase = blockIdx.x * TOK_PER_BLK;

    // per-thread staging coordinates (A tile: 64x64 floats = 1024 float4, 8 per thread)
    // per-thread issues 8 (A) + 2 (W) = 10 async b128 copies per K-tile.
    auto issue_tile = [&](int k0, int buf) {
        #pragma unroll
        for (int it = 0; it < 8; ++it) {
            const int idx = tid + it * 128;      // float4 index 0..1023
            const int r   = idx >> 4;            // 16 float4 per row
            const int c   = (idx & 15) << 2;     // column in floats
            async_cp16(xnorm + (size_t)(tokBase + r) * K_DIM + k0 + c,
                       &As[buf][r * LDS_STR + c]);
        }
        #pragma unroll
        for (int it = 0; it < 2; ++it) {
            const int idx = tid + it * 128;      // float4 index 0..255
            const int r   = idx >> 4;
            const int c   = (idx & 15) << 2;
            async_cp16(W + (size_t)r * K_DIM + k0 + c,
                       &Ws[buf][r * LDS_STR + c]);
        }
    };

    v8f acc;
    #pragma unroll
    for (int i = 0; i < 8; ++i) acc[i] = 0.0f;

    issue_tile(0, 0);

    for (int t = 0; t < N_KTILES; ++t) {
        const int cur = t & 1;
        if (t + 1 < N_KTILES) {
            issue_tile((t + 1) * BK, cur ^ 1);
            // 10 newest outstanding ops are tile t+1's; in-order completion
            // means waiting to <=10 drains all of tile t.
            asm volatile("s_wait_asynccnt 0xa" ::: "memory");
        } else {
            asm volatile("s_wait_asynccnt 0x0" ::: "memory");
        }
        __syncthreads();   // all waves' tile-t copies visible in LDS

        const float* aRow = &As[cur][(wave * TOK_PER_WAVE + row16) * LDS_STR];
        const float* bRow = &Ws[cur][row16 * LDS_STR];
        #pragma unroll
        for (int kk = 0; kk < BK; kk += 4) {
            // A 16x4 f32 frag: lane holds A[row16][kk+khalf], A[row16][kk+khalf+1]
            v2f a = *(const v2f*)(aRow + kk + khalf);
            // B 4x16 f32 frag (B[k][n] = W[n][k])
            v2f b = *(const v2f*)(bRow + kk + khalf);
            acc = __builtin_amdgcn_wmma_f32_16x16x4_f32(
                false, a, false, b, (short)0, acc, false, false);
        }
        __syncthreads();   // buffer 'cur' may be overwritten by tile t+2's issue
    }

    // C-layout: VGPR r, lanes 0-15 -> token tileBase+r, lanes 16-31 -> token tileBase+r+8,
    // N = lane&15 = flat output index m (= i*4+j of the 4x4).
    const float fac = factor[0];
    const float bi  = bias[row16];
    const int   tokTile = tokBase + wave * TOK_PER_WAVE + ((lane >> 4) << 3);

    #pragma unroll
    for (int r = 0; r < 8; ++r) {
        float h = __expf(acc[r] * fac + bi);
        #pragma unroll 1
        for (int it = 0; it < 20; ++it) {
            // row normalize: sum over j (lane bits 0,1)
            float s = h + __shfl_xor(h, 1);
            s += __shfl_xor(s, 2);
            h = h / s;
            // col normalize: sum over i (lane bits 2,3)
            s = h + __shfl_xor(h, 4);
            s += __shfl_xor(s, 8);
            h = h / s;
        }
        h_out[(size_t)(tokTile + r) * 16 + row16] = h;
    }
}

// ---------------------------------------------------------------------------
// Kernel 2: out[tok][i][:] = sum_j h[tok][i*4+j] * x[tok][j][:]
// Pure bandwidth streaming, float4 coalesced. One token per block.
// ---------------------------------------------------------------------------
__global__ __launch_bounds__(256) void MCHyper_mix_kernel(
    const float* __restrict__ x,     // (N_TOK, 4, 2048)
    const float* __restrict__ h,     // (N_TOK, 16)
    float* __restrict__ out)         // (N_TOK, 4, 2048)
{
    __shared__ float hs[16];
    const int tid = threadIdx.x;
    const size_t tok = blockIdx.x;
    if (tid < 16) hs[tid] = h[tok * 16 + tid];
    __syncthreads();

    const float4* xt = (const float4*)(x + tok * 4 * 2048);
    float4*       ot = (float4*)(out + tok * 4 * 2048);

    for (int c = tid; c < 512; c += 256) {        // 512 float4 per row of 2048
        const float4 x0 = xt[0 * 512 + c];
        const float4 x1 = xt[1 * 512 + c];
        const float4 x2 = xt[2 * 512 + c];
        const float4 x3 = xt[3 * 512 + c];
        #pragma unroll
        for (int i = 0; i < 4; ++i) {
            const float a0 = hs[4 * i + 0], a1 = hs[4 * i + 1];
            const float a2 = hs[4 * i + 2], a3 = hs[4 * i + 3];
            float4 o;
            o.x = a0 * x0.x + a1 * x1.x + a2 * x2.x + a3 * x3.x;
            o.y = a0 * x0.y + a1 * x1.y + a2 * x2.y + a3 * x3.y;
            o.z = a0 * x0.z + a1 * x1.z + a2 * x2.z + a3 * x3.z;
            o.w = a0 * x0.w + a1 * x1.w + a2 * x2.w + a3 * x3.w;
            ot[i * 512 + c] = o;
        }
    }
}

extern "C" void kernel_launch(void* const* d_in, const int* in_sizes, int n_in,
                              void* d_out, int out_size, void* d_ws, size_t ws_size,
                              hipStream_t stream) {
    const float* x      = (const float*)d_in[0];   // (2,4096,4,2048)
    const float* xnorm  = (const float*)d_in[1];   // (2,4096,8192)
    const float* W      = (const float*)d_in[2];   // (16,8192)
    const float* factor = (const float*)d_in[3];   // (1,)
    const float* bias   = (const float*)d_in[4];   // (4,4)

    float* h_ws = (float*)d_ws;                    // N_TOK*16 floats = 512 KB

    MCHyper_logits_sinkhorn_kernel<<<N_TOK / TOK_PER_BLK, 128, 0, stream>>>(
        xnorm, W, factor, bias, h_ws);
    MCHyper_mix_kernel<<<N_TOK, 256, 0, stream>>>(x, h_ws, (float*)d_out);
}